// LightGCN_17978733101580
// MI455X (gfx1250) — compile-verified
//
#include <hip/hip_runtime.h>
#include <hip/hip_bf16.h>

// LightGCN forward for MI455X (gfx1250, wave32).
//
// d_out layout (floats): [users 75000*64][items 75000*64][embs 150000*4*64]
// users||items is exactly light_out (n x 64) flattened, so light_out is the
// first n*64 floats of d_out, and the embs stack (n,4,64) is computed in
// place in the tail region: layer l lives at i*256 + l*64 + d.

typedef __attribute__((ext_vector_type(2))) float v2f;
typedef __attribute__((ext_vector_type(8))) float v8f;

// ---------------------------------------------------------------------------
// Kernel 1: embs layer 0 = concat(user_emb, item_emb); layers 1..3 = 0.
// One thread per float4 of one node row (n*16 threads).
// ---------------------------------------------------------------------------
__global__ void lightgcn_init_kernel(const float4* __restrict__ user_emb,
                                     const float4* __restrict__ item_emb,
                                     float* __restrict__ embs,
                                     int n_users, int n) {
    int t = blockIdx.x * blockDim.x + threadIdx.x;
    if (t >= n * 16) return;
    int i = t >> 4;        // node
    int q = t & 15;        // float4 chunk within 64-dim row
    float4 v = (i < n_users) ? user_emb[(size_t)i * 16 + q]
                             : item_emb[(size_t)(i - n_users) * 16 + q];
    float4* row = (float4*)(embs + (size_t)i * 256);
    float4 z = make_float4(0.f, 0.f, 0.f, 0.f);
    row[q]      = v;   // layer 0
    row[16 + q] = z;   // layer 1
    row[32 + q] = z;   // layer 2
    row[48 + q] = z;   // layer 3
}

// ---------------------------------------------------------------------------
// Kernel 2: one SpMM layer.  16 lanes cooperate on one edge: lane q handles
// 4 consecutive dims (float4), so the 256B gather/scatter per edge is fully
// coalesced.  Accumulation with f32 global atomics (L2-resident).
//   embs[row, layer, :] += val * embs[col, layer-1, :]
// ---------------------------------------------------------------------------
__global__ void lightgcn_spmm_kernel(const int* __restrict__ erow,
                                     const int* __restrict__ ecol,
                                     const float* __restrict__ eval_,
                                     float* __restrict__ embs,
                                     int n_edges, int layer) {
    int t = blockIdx.x * blockDim.x + threadIdx.x;
    int e = t >> 4;
    if (e >= n_edges) return;
    int q = t & 15;

    int   r = erow[e];
    int   c = ecol[e];
    float v = eval_[e];

    const float4* src = (const float4*)(embs + (size_t)c * 256 + (size_t)(layer - 1) * 64);
    float4 x = src[q];

    float* dst = embs + (size_t)r * 256 + (size_t)layer * 64 + (size_t)q * 4;
    atomicAdd(dst + 0, v * x.x);
    atomicAdd(dst + 1, v * x.y);
    atomicAdd(dst + 2, v * x.z);
    atomicAdd(dst + 3, v * x.w);
}

// ---------------------------------------------------------------------------
// Kernel 3: light_out = mean over the 4 layers, via V_WMMA_F32_16X16X4_F32.
// Each wave owns a 16-row x 16-dim output tile:
//   D(16x16) = sum_{l=0..3} sum_{kb=0..3} A(16x4) x B(4x16)
// where A = embs[row0:row0+16, l, j0+4*kb : j0+4*kb+4]  (data)
//       B = rows 4*kb..4*kb+3 of the 16x16 matrix 0.25*I (identity slice)
// A layout per ISA (16x4 f32): lanes 0-15 -> K0(K1), lanes 16-31 -> K2(K3),
// i.e. one aligned float2 load per lane.  D layout: VGPR v, lane j -> row
// (v + 8*(j>=16)), col (j&15).
// ---------------------------------------------------------------------------
__global__ void lightgcn_mean_wmma_kernel(const float* __restrict__ embs,
                                          float* __restrict__ light_out,
                                          int n) {
    int wavesPerBlock = blockDim.x >> 5;
    int wave = blockIdx.x * wavesPerBlock + (threadIdx.x >> 5);
    int lane = threadIdx.x & 31;

    int nRowTiles = n >> 4;              // n is a multiple of 16
    int rowTile = wave >> 2;             // 16 rows per tile
    int colTile = wave & 3;              // 4 tiles of 16 dims
    if (rowTile >= nRowTiles) return;    // wave-uniform: EXEC stays all-ones

    int row0 = rowTile << 4;
    int j0   = colTile << 4;
    int half = lane >> 4;                // 0: lanes 0-15, 1: lanes 16-31
    int lj   = lane & 15;
    int kk   = half * 2;                 // K pair handled by this lane half
    int arow = row0 + lj;

    v8f c = {0.f, 0.f, 0.f, 0.f, 0.f, 0.f, 0.f, 0.f};

#pragma unroll
    for (int l = 0; l < 4; ++l) {
#pragma unroll
        for (int kb = 0; kb < 4; ++kb) {
            // A: float2 = embs[arow, l, j0 + 4*kb + kk .. +1]
            const v2f a = *(const v2f*)(embs + (size_t)arow * 256
                                             + (size_t)l * 64
                                             + j0 + kb * 4 + kk);
            // B: 0.25 * identity slice (rows 4*kb+kk, 4*kb+kk+1; col lj)
            v2f b;
            b.x = (lj == kb * 4 + kk)     ? 0.25f : 0.0f;
            b.y = (lj == kb * 4 + kk + 1) ? 0.25f : 0.0f;

            c = __builtin_amdgcn_wmma_f32_16x16x4_f32(
                    /*neg_a=*/false, a, /*neg_b=*/false, b,
                    /*c_mod=*/(short)0, c, /*reuse_a=*/false, /*reuse_b=*/false);
        }
    }

    // Store D: lane j holds rows (v + 8*half), column j0 + lj.
    float* outCol = light_out + j0 + lj;
    int rbase = row0 + half * 8;
#pragma unroll
    for (int v = 0; v < 8; ++v) {
        outCol[(size_t)(rbase + v) * 64] = c[v];
    }
}

// ---------------------------------------------------------------------------
extern "C" void kernel_launch(void* const* d_in, const int* in_sizes, int n_in,
                              void* d_out, int out_size, void* d_ws, size_t ws_size,
                              hipStream_t stream) {
    const float* user_emb = (const float*)d_in[0];
    const float* item_emb = (const float*)d_in[1];
    const int*   edge_row = (const int*)d_in[2];
    const int*   edge_col = (const int*)d_in[3];
    const float* edge_val = (const float*)d_in[4];

    const int D       = 64;
    int n_users = in_sizes[0] / D;
    int n_items = in_sizes[1] / D;
    int n_edges = in_sizes[2];
    int n       = n_users + n_items;

    float* light_out = (float*)d_out;                       // n*64 (users||items)
    float* embs      = light_out + (size_t)n * D;           // n*4*64, in place

    // Layer 0 + zero-init of layers 1..3 (re-done every call: deterministic).
    {
        int threads = n * 16;
        int blocks  = (threads + 255) / 256;
        lightgcn_init_kernel<<<blocks, 256, 0, stream>>>(
            (const float4*)user_emb, (const float4*)item_emb, embs, n_users, n);
    }

    // 3 propagation layers; stream order gives the layer dependency.
    {
        long long threads = (long long)n_edges * 16;
        int blocks = (int)((threads + 255) / 256);
        for (int layer = 1; layer <= 3; ++layer) {
            lightgcn_spmm_kernel<<<blocks, 256, 0, stream>>>(
                edge_row, edge_col, edge_val, embs, n_edges, layer);
        }
    }

    // Mean over the 4 layers via WMMA; writes users||items region.
    {
        int waves  = (n / 16) * 4;       // one wave per 16x16 output tile
        int blocks = (waves + 7) / 8;    // 8 waves per 256-thread block
        lightgcn_mean_wmma_kernel<<<blocks, 256, 0, stream>>>(embs, light_out, n);
    }
}